// GaussianBlur_73847667687780
// MI455X (gfx1250) — compile-verified
//
#include <hip/hip_runtime.h>
#include <hip/hip_bf16.h>
#include <math.h>

// ---------------------------------------------------------------------------
// Per-batch separable Gaussian blur (K=161, reflect pad) on MI455X (gfx1250).
//
// Each separable pass is a tiled fp32 GEMM on V_WMMA_F32_16X16X4_F32:
//   D[16x16] += A[16x4] * B[4x16],  K = 176 = 44*4
//   vertical :  A[m,k] = w[k-m]  (Toeplitz, registers),  B[k,n] = strip[k][n]
//   horizontal: A[m,k] = strip[m][k],  B[k,n] = w[k-n]  (Toeplitz, registers)
// The Toeplitz weight operand is tile-invariant -> hoisted into 44 v2f regs.
// Taps are zero-padded in LDS so fragment loads are branch-free (EXEC all-1s).
// Image strips live in LDS ([16][674] f32, ~43 KB) so HBM traffic is exactly
// one read + one write per pass; the data operand pair is contiguous and
// 8B-aligned -> one ds_load_b64 (immediate offset) + one v_wmma per chunk.
// Horizontal pass is in place on d_out (each WG owns its 16 rows; the LDS
// stage + barrier precedes any store).
// ---------------------------------------------------------------------------

#define BATCH   64
#define CH      3
#define HH      512
#define WW      512
#define RADIUS  80
#define KSZ     161             // taps
#define PADH    (HH + 2*RADIUS) // 672 padded extent
#define KTOT    176             // 16 + 160 band depth
#define KCHUNKS (KTOT / 4)      // 44
#define SPITCH  674             // LDS pitch: even (b64 align) + conflict-free
#define WPAD    192             // 16 zeros | 161 taps | 15 zeros

typedef __attribute__((ext_vector_type(2))) float v2f;
typedef __attribute__((ext_vector_type(8))) float v8f;

__device__ __forceinline__ int reflect_idx(int i, int n) {
    // jnp.pad mode="reflect": -1 -> 1, n -> n-2 (R < n, single reflection)
    i = (i < 0) ? -i : i;
    i = (i >= n) ? (2 * n - 2 - i) : i;
    return i;
}

// ---------------------------------------------------------------------------
// Kernel 1: per-batch normalized Gaussian taps  w[b][k], k in [0,161)
// ---------------------------------------------------------------------------
__global__ __launch_bounds__(192) void gb_weights_kernel(
    const float* __restrict__ sigmas,   // [200]
    const int*   __restrict__ steps,    // [64]
    float*       __restrict__ w)        // [64][161]
{
    __shared__ float s_pdf[KSZ];
    const int b = blockIdx.x;
    const int k = threadIdx.x;
    const float sigma = sigmas[steps[b]] + 1e-6f;
    if (k < KSZ) {
        const float t = (float)(k - RADIUS) / sigma;
        s_pdf[k] = __expf(-0.5f * t * t);
    }
    __syncthreads();
    if (k < KSZ) {
        float s = 0.f;
        #pragma unroll 1
        for (int i = 0; i < KSZ; ++i) s += s_pdf[i];
        w[b * KSZ + k] = s_pdf[k] / s;
    }
}

// ---------------------------------------------------------------------------
// Kernel 2: vertical pass.  One WG per (image, 16-col strip).
//   grid = B*C*(W/16) = 6144,  block = 256 (8 waves, 4 tiles each)
// ---------------------------------------------------------------------------
__global__ __launch_bounds__(256) void gb_vert_kernel(
    const float* __restrict__ x,        // [B*C][H][W]
    const float* __restrict__ w,        // [B][161]
    float*       __restrict__ y)        // [B*C][H][W]
{
    __shared__ float s_strip[16 * SPITCH];  // [col][padded row], pitch 674
    __shared__ float s_w[WPAD];             // zero-padded taps at +16

    const int bc  = blockIdx.x >> 5;          // image index (b*C + c)
    const int c0  = (blockIdx.x & 31) << 4;   // column tile origin
    const int b   = bc / CH;
    const int tid = threadIdx.x;

    const float* src = x + (size_t)bc * (HH * WW);
    float*       dst = y + (size_t)bc * (HH * WW);

    // stage zero-padded taps (branch only here, never in the WMMA loop)
    for (int i = tid; i < WPAD; i += 256) {
        const int k = i - 16;
        s_w[i] = ((unsigned)k < (unsigned)KSZ) ? w[b * KSZ + k] : 0.f;
    }
    // stage 672x16 input strip, transposed to [col][row] (64B per 16 lanes)
    for (int e = tid; e < PADH * 16; e += 256) {
        const int r   = e >> 4;
        const int col = e & 15;
        const int ir  = reflect_idx(r - RADIUS, HH);
        s_strip[col * SPITCH + r] = src[ir * WW + c0 + col];
    }
    __syncthreads();

    const int wave = tid >> 5;
    const int lane = tid & 31;
    const int ln   = lane & 15;       // = M for A, = N for B/D
    const int half = lane >> 4;
    const int koff = half << 1;       // f32 16x4 layout: lanes 16-31 hold K+2

    // Hoist A (Toeplitz weights): afrag[kc] = { w[4kc+koff-ln], w[4kc+koff+1-ln] }
    v2f afrag[KCHUNKS];
    {
        const float* wp = &s_w[16 + koff - ln];
        #pragma unroll
        for (int kc = 0; kc < KCHUNKS; ++kc) {
            v2f a; a.x = wp[4 * kc]; a.y = wp[4 * kc + 1];
            afrag[kc] = a;
        }
    }

    #pragma unroll 1
    for (int t4 = 0; t4 < 4; ++t4) {
        const int tt  = wave * 4 + t4;    // tile index along rows, 0..31
        const int r0t = tt << 4;
        // B base: strip[col=ln][row = r0t + koff + k], contiguous 8B pairs
        const float* bp = &s_strip[ln * SPITCH + r0t + koff];
        v8f acc = {0.f, 0.f, 0.f, 0.f, 0.f, 0.f, 0.f, 0.f};

        #pragma unroll
        for (int kc = 0; kc < KCHUNKS; ++kc) {
            const v2f bb = *(const v2f*)(bp + 4 * kc);   // ds_load_b64
            acc = __builtin_amdgcn_wmma_f32_16x16x4_f32(
                false, afrag[kc], false, bb, (short)0, acc, false, false);
        }

        // D layout: VGPR v -> row M = v + 8*half, lanes give N = ln
        #pragma unroll
        for (int v = 0; v < 8; ++v) {
            const int row = r0t + v + (half << 3);
            dst[row * WW + c0 + ln] = acc[v];
        }
    }
}

// ---------------------------------------------------------------------------
// Kernel 3: horizontal pass, in place on y.  One WG per (image, 16-row strip).
//   grid = B*C*(H/16) = 6144,  block = 256
// ---------------------------------------------------------------------------
__global__ __launch_bounds__(256) void gb_horiz_kernel(
    float*       __restrict__ y,        // [B*C][H][W], read then overwritten
    const float* __restrict__ w)        // [B][161]
{
    __shared__ float s_strip[16 * SPITCH];  // [row][padded col], pitch 674
    __shared__ float s_w[WPAD];

    const int bc  = blockIdx.x >> 5;
    const int r0  = (blockIdx.x & 31) << 4;   // row tile origin
    const int b   = bc / CH;
    const int tid = threadIdx.x;

    float* img = y + (size_t)bc * (HH * WW);

    for (int i = tid; i < WPAD; i += 256) {
        const int k = i - 16;
        s_w[i] = ((unsigned)k < (unsigned)KSZ) ? w[b * KSZ + k] : 0.f;
    }
    // stage 16x672 strip (contiguous along width)
    for (int e = tid; e < 16 * PADH; e += 256) {
        const int row = e / PADH;
        const int i   = e - row * PADH;
        const int ic  = reflect_idx(i - RADIUS, WW);
        s_strip[row * SPITCH + i] = img[(r0 + row) * WW + ic];
    }
    __syncthreads();   // all reads of the owned rows complete before any store

    const int wave = tid >> 5;
    const int lane = tid & 31;
    const int ln   = lane & 15;
    const int half = lane >> 4;
    const int koff = half << 1;

    // Hoist B (Toeplitz weights): bfrag[kc] = { w[4kc+koff-ln], w[4kc+koff+1-ln] }
    v2f bfrag[KCHUNKS];
    {
        const float* wp = &s_w[16 + koff - ln];
        #pragma unroll
        for (int kc = 0; kc < KCHUNKS; ++kc) {
            v2f bb; bb.x = wp[4 * kc]; bb.y = wp[4 * kc + 1];
            bfrag[kc] = bb;
        }
    }

    #pragma unroll 1
    for (int t4 = 0; t4 < 4; ++t4) {
        const int tt  = wave * 4 + t4;    // tile index along columns
        const int c0t = tt << 4;
        // A base: strip[row=ln][col = c0t + koff + k], contiguous 8B pairs
        const float* ap = &s_strip[ln * SPITCH + c0t + koff];
        v8f acc = {0.f, 0.f, 0.f, 0.f, 0.f, 0.f, 0.f, 0.f};

        #pragma unroll
        for (int kc = 0; kc < KCHUNKS; ++kc) {
            const v2f aa = *(const v2f*)(ap + 4 * kc);   // ds_load_b64
            acc = __builtin_amdgcn_wmma_f32_16x16x4_f32(
                false, aa, false, bfrag[kc], (short)0, acc, false, false);
        }

        #pragma unroll
        for (int v = 0; v < 8; ++v) {
            const int row = r0 + v + (half << 3);
            img[row * WW + c0t + ln] = acc[v];
        }
    }
}

// ---------------------------------------------------------------------------
extern "C" void kernel_launch(void* const* d_in, const int* in_sizes, int n_in,
                              void* d_out, int out_size, void* d_ws, size_t ws_size,
                              hipStream_t stream) {
    const float* x      = (const float*)d_in[0];   // [64,3,512,512]
    const float* sigmas = (const float*)d_in[1];   // [200]
    const int*   steps  = (const int*)d_in[2];     // [64]
    float*       out    = (float*)d_out;           // [64,3,512,512]
    float*       wts    = (float*)d_ws;            // [64][161] scratch

    (void)in_sizes; (void)n_in; (void)out_size; (void)ws_size;

    gb_weights_kernel<<<BATCH, 192, 0, stream>>>(sigmas, steps, wts);

    const int nstrips = BATCH * CH * (WW / 16);    // 6144
    gb_vert_kernel <<<nstrips, 256, 0, stream>>>(x, wts, out);
    gb_horiz_kernel<<<nstrips, 256, 0, stream>>>(out, wts);
}